// VectorQuantizer_41472204210905
// MI455X (gfx1250) — compile-verified
//
#include <hip/hip_runtime.h>

typedef __attribute__((ext_vector_type(2))) float v2f;
typedef __attribute__((ext_vector_type(8))) float v8f;
typedef unsigned int u32x4 __attribute__((ext_vector_type(4)));
typedef int          i32x8 __attribute__((ext_vector_type(8)));
typedef int          i32x4 __attribute__((ext_vector_type(4)));

#define K_CODES 8192
#define D_DIM   256
#define N_VEC   32768
#define HW      4096
#define ZQ_ELEMS 8388608   // 8*256*64*64

// ---------------------------------------------------------------------------
// Kernel 1: per-code squared norms ||e_k||^2
// ---------------------------------------------------------------------------
__global__ __launch_bounds__(256)
void vq_norms(const float* __restrict__ cb, float* __restrict__ cnorm) {
    __shared__ float red[256];
    const int k = blockIdx.x;
    const float v = cb[(size_t)k * D_DIM + threadIdx.x];
    red[threadIdx.x] = v * v;
    __syncthreads();
    for (int s = 128; s > 0; s >>= 1) {
        if ((int)threadIdx.x < s) red[threadIdx.x] += red[threadIdx.x + s];
        __syncthreads();
    }
    if (threadIdx.x == 0) cnorm[k] = red[0];
}

// ---------------------------------------------------------------------------
// Kernel 2: WMMA distance GEMM + running argmin
//   block = 256 threads (8 waves), owns 64 rows (4 x 16-row tiles)
//   A tile staged to LDS via Tensor Data Mover (d-major, padded line stride 65)
//   wave w scans codes [w*1024, w*1024+1024)
// ---------------------------------------------------------------------------
#define MROWS 64
#define RT    4
#define LDA   65    // dwords per d-line after TDM padding (64 data + 1 pad)

__global__ __launch_bounds__(256)
void vq_argmin(const float* __restrict__ z, const float* __restrict__ cb,
               const float* __restrict__ cnorm,
               int* __restrict__ out_idx, float* __restrict__ out_idx_f) {
    __shared__ float As[D_DIM * LDA];     // As[d*65 + row], 66,560 B
    __shared__ float rmin_s[8][MROWS];
    __shared__ int   ridx_s[8][MROWS];

    const int t    = threadIdx.x;
    const int wave = t >> 5;
    const int lane = t & 31;
    const int col  = lane & 15;
    const int half = lane >> 4;
    const int n0   = blockIdx.x * MROWS;

    // z is BCHW: element (n, d) lives at (n>>12)*(D*HW) + d*HW + (n&4095)
    const size_t zbase = (size_t)(n0 >> 12) * ((size_t)D_DIM * HW) + (size_t)(n0 & (HW - 1));

    // -----------------------------------------------------------------------
    // Stage A tile with the Tensor Data Mover:
    //   memory tile: dim0 = 64 contiguous rows (x), dim1 = 256 channels (y),
    //   tensor_dim0_stride = HW.  pad 1 dword after every 64 -> LDS stride 65.
    // -----------------------------------------------------------------------
    if (t < 32) {                       // one wave issues the DMA
        const unsigned long long ga = (unsigned long long)(uintptr_t)(z + zbase);
        const unsigned int lds_off  = (unsigned int)(uintptr_t)(&As[0]);

        u32x4 g0;
        g0[0] = 1u;                                   // count=1, user descriptor
        g0[1] = lds_off;                              // lds_addr (bytes)
        g0[2] = (unsigned int)(ga & 0xFFFFFFFFull);   // global_addr[31:0]
        g0[3] = (unsigned int)((ga >> 32) & 0x1FFFFFFull) | 0x80000000u; // [56:32], type=2

        i32x8 g1;
        g1[0] = (int)((2u << 16)        //  data_size = 4 bytes
                    | (1u << 20)        //  pad_enable
                    | (5u << 22));      //  pad_interval: every 64 dwords (amount=1 dword)
        g1[1] = (int)(4096u << 16);     //  tensor_dim0 = 4096 (low 16 into [31:16])
        g1[2] = (int)(256u << 16);      //  tensor_dim0 hi=0 | tensor_dim1 = 256
        g1[3] = (int)(64u << 16);       //  tensor_dim1 hi=0 | tile_dim0 = 64
        g1[4] = 256;                    //  tile_dim1 = 256, tile_dim2 = 0
        g1[5] = 4096;                   //  tensor_dim0_stride = HW
        g1[6] = 0;                      //  stride hi / tensor_dim1_stride lo
        g1[7] = 0;

        i32x4 gz4; gz4[0] = 0; gz4[1] = 0; gz4[2] = 0; gz4[3] = 0;
        i32x8 gz8;
        #pragma unroll
        for (int i = 0; i < 8; ++i) gz8[i] = 0;

        __builtin_amdgcn_tensor_load_to_lds(g0, g1, gz4, gz4, gz8, 0);
        __builtin_amdgcn_s_wait_tensorcnt(0);
    }
    __syncthreads();

    float rmin[RT][8];
    int   ridx[RT][8];
    #pragma unroll
    for (int rt = 0; rt < RT; ++rt)
        #pragma unroll
        for (int i = 0; i < 8; ++i) { rmin[rt][i] = 3.0e38f; ridx[rt][i] = 0x7fffffff; }

    const int kbeg = wave * (K_CODES / 8);
    const int kend = kbeg + (K_CODES / 8);

    for (int k0 = kbeg; k0 < kend; k0 += 16) {
        v8f acc[RT];
        #pragma unroll
        for (int rt = 0; rt < RT; ++rt) acc[rt] = (v8f){0.f,0.f,0.f,0.f,0.f,0.f,0.f,0.f};

        const float* brow = cb + (size_t)(k0 + col) * D_DIM + 2 * half;
        if (k0 + 16 < kend)             // prefetch next k-tile's codebook rows
            __builtin_prefetch(brow + D_DIM * 16, 0, 1);

        #pragma unroll 4
        for (int d = 0; d < D_DIM; d += 4) {
            // B fragment (4x16): lanes 0-15 rows k=d,d+1; lanes 16-31 rows d+2,d+3
            v2f bfrag = *(const v2f*)(brow + d);
            const int dl = d + 2 * half;
            #pragma unroll
            for (int rt = 0; rt < RT; ++rt) {
                // A fragment (16x4) from d-major LDS (TDM layout)
                const int arow = rt * 16 + col;
                v2f afrag;
                afrag.x = As[dl * LDA + arow];
                afrag.y = As[(dl + 1) * LDA + arow];
                acc[rt] = __builtin_amdgcn_wmma_f32_16x16x4_f32(
                    false, afrag, false, bfrag, (short)0, acc[rt], false, false);
            }
        }

        const float cn = cnorm[k0 + col];
        #pragma unroll
        for (int rt = 0; rt < RT; ++rt)
            #pragma unroll
            for (int i = 0; i < 8; ++i) {
                float dist = cn - 2.0f * acc[rt][i];
                if (dist < rmin[rt][i]) { rmin[rt][i] = dist; ridx[rt][i] = k0 + col; }
            }
    }

    // Reduce across the 16 columns (xor masks 1..8 stay inside each 16-lane half)
    #pragma unroll
    for (int rt = 0; rt < RT; ++rt) {
        #pragma unroll
        for (int i = 0; i < 8; ++i) {
            float v = rmin[rt][i];
            int   ix = ridx[rt][i];
            #pragma unroll
            for (int m = 1; m < 16; m <<= 1) {
                float ov = __shfl_xor(v, m, 32);
                int  oix = __shfl_xor(ix, m, 32);
                if (ov < v || (ov == v && oix < ix)) { v = ov; ix = oix; }
            }
            if (col == 0) {  // lanes 0 (rows i) and 16 (rows i+8)
                const int row = rt * 16 + i + 8 * half;
                rmin_s[wave][row] = v;
                ridx_s[wave][row] = ix;
            }
        }
    }
    __syncthreads();

    // Cross-wave reduce + emit
    if (t < MROWS) {
        float best = rmin_s[0][t];
        int   bix  = ridx_s[0][t];
        #pragma unroll
        for (int w = 1; w < 8; ++w) {
            float v = rmin_s[w][t];
            int  ix = ridx_s[w][t];
            if (v < best || (v == best && ix < bix)) { best = v; bix = ix; }
        }
        out_idx[n0 + t]   = bix;
        out_idx_f[n0 + t] = (float)bix;
    }
}

// ---------------------------------------------------------------------------
// Kernel 3: gather z_q (BCHW) + per-block partial sum of (z_q - z)^2
// ---------------------------------------------------------------------------
__global__ __launch_bounds__(256)
void vq_gather(const float* __restrict__ z, const float* __restrict__ cb,
               const int* __restrict__ idx, float* __restrict__ zq,
               float* __restrict__ partial) {
    __shared__ float red[256];
    const size_t g  = (size_t)blockIdx.x * 256 + threadIdx.x;  // BCHW-linear
    const int b   = (int)(g >> 20);          // C*HW = 2^20
    const int rem = (int)(g & 0xFFFFF);
    const int c   = rem >> 12;               // HW = 2^12
    const int hw  = rem & (HW - 1);
    const int n   = (b << 12) | hw;

    const float q  = cb[(size_t)idx[n] * D_DIM + c];
    const float zv = z[g];
    zq[g] = q;
    const float d = q - zv;
    red[threadIdx.x] = d * d;
    __syncthreads();
    for (int s = 128; s > 0; s >>= 1) {
        if ((int)threadIdx.x < s) red[threadIdx.x] += red[threadIdx.x + s];
        __syncthreads();
    }
    if (threadIdx.x == 0) partial[blockIdx.x] = red[0];
}

// ---------------------------------------------------------------------------
// Kernel 4: deterministic final loss reduction
// ---------------------------------------------------------------------------
__global__ __launch_bounds__(256)
void vq_loss(const float* __restrict__ partial, float* __restrict__ loss_out) {
    __shared__ float red[256];
    float s = 0.0f;
    for (int i = threadIdx.x; i < ZQ_ELEMS / 256; i += 256) s += partial[i];
    red[threadIdx.x] = s;
    __syncthreads();
    for (int st = 128; st > 0; st >>= 1) {
        if ((int)threadIdx.x < st) red[threadIdx.x] += red[threadIdx.x + st];
        __syncthreads();
    }
    if (threadIdx.x == 0) *loss_out = 1.25f * red[0] / (float)ZQ_ELEMS;
}

// ---------------------------------------------------------------------------
extern "C" void kernel_launch(void* const* d_in, const int* in_sizes, int n_in,
                              void* d_out, int out_size, void* d_ws, size_t ws_size,
                              hipStream_t stream) {
    const float* z  = (const float*)d_in[0];   // [8,256,64,64]
    const float* cb = (const float*)d_in[1];   // [8192,256]

    float* out  = (float*)d_out;
    float* zq   = out;                    // 8388608 floats (BCHW)
    float* loss = out + ZQ_ELEMS;         // 1 float
    float* idxf = out + ZQ_ELEMS + 1;     // 32768 floats (indices as fp32)

    char*  ws      = (char*)d_ws;
    float* cnorm   = (float*)ws;                                    //  8192 f32
    int*   idx     = (int*)(ws + (size_t)K_CODES * 4);              // 32768 i32
    float* partial = (float*)(ws + (size_t)K_CODES * 4
                                 + (size_t)N_VEC * 4);              // 32768 f32

    vq_norms <<<K_CODES, 256, 0, stream>>>(cb, cnorm);
    vq_argmin<<<N_VEC / MROWS, 256, 0, stream>>>(z, cb, cnorm, idx, idxf);
    vq_gather<<<ZQ_ELEMS / 256, 256, 0, stream>>>(z, cb, idx, zq, partial);
    vq_loss  <<<1, 256, 0, stream>>>(partial, loss);
}